// Decorrelation_72086731096883
// MI455X (gfx1250) — compile-verified
//
#include <hip/hip_runtime.h>
#include <hip/hip_bf16.h>

typedef __attribute__((ext_vector_type(2))) float v2f;
typedef __attribute__((ext_vector_type(8))) float v8f;

#define DIM   16
#define KC    11                 // number of basis functions per dim
#define KEXT  (DIM * KC + DIM)   // 192: 176 spline columns + 16 identity columns
#define NK4   (KEXT / 4)         // 48 WMMA K-steps
#define NSEG  8                  // KC - SPLINE_DEG
#define WSTRIDE 196              // 196%4==0 (16B-aligned rows), 196%64==4 (conflict-free)
#define WAVES_PER_BLOCK 4
#define BLOCK (WAVES_PER_BLOCK * 32)
#define ROWS_PER_BLOCK (WAVES_PER_BLOCK * 16)

// out[n,v] = sum_k W[n,k] * P[k,v] where
//   W[n, c*11+k] = basis(xc[n,c])_k * x[n,c]   (c = 0..15, k = 0..10)
//   W[n, 176+j]  = x[n,j]                      (identity columns)
//   P[(c,k), v]  = params[k, v*(v-1)/2 + c] for c < v else 0
//   P[176+j, v]  = (j == v)
// Uniform-knot cubic B-spline: only 4 of 11 basis values are nonzero; compute
// them in closed form and scatter into a pre-zeroed W row.
// GEMM: 16x16 output tile per wave, K=192 via 48x v_wmma_f32_16x16x4_f32.
__global__ __launch_bounds__(BLOCK) void decor_kernel(
    const float* __restrict__ x, const float* __restrict__ params,
    const float* __restrict__ prange, float* __restrict__ out) {
    __shared__ float PT[DIM][WSTRIDE];                 // P transposed: PT[v][k]
    __shared__ float W[WAVES_PER_BLOCK][16][WSTRIDE];  // per-wave A staging
    __shared__ float lowS[DIM], xmaxS[DIM], hS[DIM], invhS[DIM];

    const int tid = threadIdx.x;

    // Build PT (constant B-matrix of the GEMM) cooperatively.
    for (int idx = tid; idx < DIM * KEXT; idx += BLOCK) {
        int v = idx / KEXT;
        int k = idx - v * KEXT;
        float val = 0.0f;
        if (k < DIM * KC) {
            int c  = k / KC;
            int kk = k - c * KC;
            if (c < v) val = params[kk * (DIM * (DIM - 1) / 2) + (v * (v - 1)) / 2 + c];
        } else {
            val = ((k - DIM * KC) == v) ? 1.0f : 0.0f;
        }
        PT[v][k] = val;
    }
    if (tid < DIM) {
        float lo0 = prange[tid];
        float hi0 = prange[DIM + tid];
        float span = hi0 - lo0;
        float lo = lo0 - 0.1f * span;
        float hi = hi0 + 0.1f * span;
        float h  = (hi - lo) / (float)NSEG;
        lowS[tid]   = lo;
        hS[tid]     = h;
        invhS[tid]  = 1.0f / h;
        xmaxS[tid]  = hi - 1e-6f * (hi - lo);   // clamp upper bound (high - eps)
    }
    __syncthreads();

    const int wave  = tid >> 5;
    const int lane  = tid & 31;
    const int row16 = lane & 15;
    const int half  = lane >> 4;   // 0: lanes 0-15, 1: lanes 16-31
    const int rowBase = blockIdx.x * ROWS_PER_BLOCK + wave * 16;
    const int grow = rowBase + row16;

    float* __restrict__ wrow = &W[wave][row16][0];

    // Zero this lane's spline-column region (exactly covers its scatter range:
    // half 0 -> cols 0..87, half 1 -> cols 88..175). Lane-private => same-wave
    // in-order LDS semantics make the later scatter stores safe w/o barrier.
    {
        float4 z = {0.0f, 0.0f, 0.0f, 0.0f};
        float4* zp = reinterpret_cast<float4*>(wrow + half * 88);
#pragma unroll
        for (int i = 0; i < 22; ++i) zp[i] = z;
    }

    // Phase 1: closed-form cubic basis, 8 dims per lane-half.
    const int cBase = half * 8;
    const float4* xrow = reinterpret_cast<const float4*>(x + (size_t)grow * DIM + cBase);
    float4 xa = xrow[0], xb = xrow[1];
    float xv8[8] = {xa.x, xa.y, xa.z, xa.w, xb.x, xb.y, xb.z, xb.w};
#pragma unroll
    for (int cc = 0; cc < 8; ++cc) {
        int c = cBase + cc;
        float xv = xv8[cc];
        float lo = lowS[c];
        float xc = fminf(fmaxf(xv, lo), xmaxS[c]);
        float uu = (xc - lo) * invhS[c];            // in [0, 8)
        float sf = fminf(floorf(uu), 7.0f);
        float u  = uu - sf;                         // fractional position in segment
        int   s  = (int)sf;
        float um = 1.0f - u;
        float u2 = u * u;
        float u3 = u2 * u;
        float um3 = um * um * um;
        float b0 = um3 * (1.0f / 6.0f);                                  // (1-u)^3/6
        float b1 = fmaf(0.5f, u3, (2.0f / 3.0f) - u2);                   // u^3/2 - u^2 + 2/3
        float b2 = fmaf(-0.5f, u3, fmaf(0.5f, u2, fmaf(0.5f, u, 1.0f / 6.0f)));
        float b3 = u3 * (1.0f / 6.0f);                                   // u^3/6
        int off = c * KC + s;
        wrow[off]     = b0 * xv;
        wrow[off + 1] = b1 * xv;
        wrow[off + 2] = b2 * xv;
        wrow[off + 3] = b3 * xv;
        wrow[DIM * KC + c] = xv;                    // identity column
    }
    __syncthreads();

    // Phase 2: K=192 GEMM. A layout (32-bit 16x4): lanes 0-15 hold K={k0,k0+1},
    // lanes 16-31 hold K={k0+2,k0+3} for row lane&15; B mirrors with row<->col.
    v8f acc = {};
    const int e = half * 2;
#pragma unroll
    for (int kk = 0; kk < NK4; ++kk) {
        int k0 = kk * 4 + e;
        v2f a, b;
        a[0] = W[wave][row16][k0];
        a[1] = W[wave][row16][k0 + 1];
        b[0] = PT[row16][k0];
        b[1] = PT[row16][k0 + 1];
        acc = __builtin_amdgcn_wmma_f32_16x16x4_f32(
            /*neg_a=*/false, a, /*neg_b=*/false, b,
            /*c_mod=*/(short)0, acc, /*reuse_a=*/false, /*reuse_b=*/false);
    }

    // C/D layout (32-bit 16x16): VGPR i -> row i + 8*half, col = lane&15.
#pragma unroll
    for (int i = 0; i < 8; ++i)
        out[(size_t)(rowBase + i + half * 8) * DIM + row16] = acc[i];
}

// Penalty scalars over params (11 x 120): second-order, first-order, sum-of-squares.
__global__ __launch_bounds__(128) void pen_kernel(const float* __restrict__ p,
                                                  float* __restrict__ out3) {
    __shared__ float s2[128], s1[128], s0[128];
    const int t = threadIdx.x;
    const int L = DIM * (DIM - 1) / 2;  // 120
    float a2 = 0.0f, a1 = 0.0f, a0 = 0.0f;
    for (int l = t; l < L; l += 128) {
        float col[KC];
#pragma unroll
        for (int k = 0; k < KC; ++k) col[k] = p[k * L + l];
#pragma unroll
        for (int k = 0; k < KC; ++k) a0 += col[k] * col[k];
#pragma unroll
        for (int k = 0; k < KC - 1; ++k) { float d = col[k + 1] - col[k]; a1 += d * d; }
#pragma unroll
        for (int k = 0; k < KC - 2; ++k) {
            float d = col[k + 2] - 2.0f * col[k + 1] + col[k];
            a2 += d * d;
        }
    }
    s2[t] = a2; s1[t] = a1; s0[t] = a0;
    __syncthreads();
    for (int off = 64; off > 0; off >>= 1) {
        if (t < off) { s2[t] += s2[t + off]; s1[t] += s1[t + off]; s0[t] += s0[t + off]; }
        __syncthreads();
    }
    if (t == 0) { out3[0] = s2[0]; out3[1] = s1[0]; out3[2] = s0[0]; }
}

extern "C" void kernel_launch(void* const* d_in, const int* in_sizes, int n_in,
                              void* d_out, int out_size, void* d_ws, size_t ws_size,
                              hipStream_t stream) {
    const float* x      = (const float*)d_in[0];   // (N, 16)
    const float* params = (const float*)d_in[1];   // (11, 120)
    const float* pr     = (const float*)d_in[2];   // (2, 16)
    float* out = (float*)d_out;                    // N*16 floats + 3 penalty scalars

    const int N = in_sizes[0] / DIM;               // 262144
    const int blocks = N / ROWS_PER_BLOCK;         // 4096
    decor_kernel<<<blocks, BLOCK, 0, stream>>>(x, params, pr, out);
    pen_kernel<<<1, 128, 0, stream>>>(params, out + (size_t)N * DIM);
}